// DigitCaps_38500086842081
// MI455X (gfx1250) — compile-verified
//
#include <hip/hip_runtime.h>
#include <hip/hip_bf16.h>
#include <math.h>

// ---------------------------------------------------------------------------
// DigitCaps dynamic routing, fused for MI455X (gfx1250, wave32, WMMA f32).
//   x: [64, 6272, 8] f32      W: [10, 6272, 8, 16] f32   out: [64,10,1,16] f32
// u_hat is never materialized: x+W (45MB) live in the 192MB L2 and u_hat is
// recomputed via V_WMMA_F32_16X16X4_F32 each time it is needed.
// s[b,c,o] = sum_{n,i} (c_ij[b,n] * x[b,n,i]) * W[c,n,i,o]  -> one long-K GEMM
// per caps (M=64,N=16,K=50176), row-scaling folded into the A fragment.
// ---------------------------------------------------------------------------

#define BATCH 64
#define CAPS  10
#define NODES 6272
#define CIN   8
#define COUT  16
#define NCHUNK 49                       // node chunks per caps
#define NPC   (NODES / NCHUNK)          // 128 nodes per chunk

typedef float v2f __attribute__((ext_vector_type(2)));
typedef float v8f __attribute__((ext_vector_type(8)));

// ---------------- init: logits = 0, mz = (0, N), s = 0 ----------------------
__global__ void __launch_bounds__(256)
caps_init_kernel(float* __restrict__ logits, float* __restrict__ mz,
                 float* __restrict__ sbuf) {
    const size_t NL = (size_t)BATCH * CAPS * NODES;
    size_t i = (size_t)blockIdx.x * 256 + threadIdx.x;
    for (size_t k = i; k < NL; k += (size_t)gridDim.x * 256)
        logits[k] = 0.0f;
    if (i < BATCH * CAPS) {
        mz[2 * i + 0] = 0.0f;
        mz[2 * i + 1] = (float)NODES;   // softmax of zeros -> weight 1/NODES
    }
    if (i < BATCH * CAPS * COUT) sbuf[i] = 0.0f;
}

__global__ void __launch_bounds__(256)
caps_zero_s_kernel(float* __restrict__ sbuf) {
    int i = blockIdx.x * 256 + threadIdx.x;
    if (i < BATCH * CAPS * COUT) sbuf[i] = 0.0f;
}

// ---------------- s_j GEMM: long-K WMMA with routing weights on A ----------
__global__ void __launch_bounds__(128)
caps_s_gemm_kernel(const float* __restrict__ x, const float* __restrict__ W,
                   const float* __restrict__ logits, const float* __restrict__ mz,
                   float* __restrict__ sbuf) {
    const int c     = blockIdx.x;        // caps index
    const int chunk = blockIdx.y;        // node chunk
    const int wave  = threadIdx.x >> 5;  // 4 waves = 4 b-tiles (M=64)
    const int lane  = threadIdx.x & 31;
    const int half  = lane >> 4;         // 0: lanes 0-15, 1: lanes 16-31
    const int l16   = lane & 15;
    const int b0    = wave * 16;
    const int b     = b0 + l16;          // A-fragment row for this lane

    const float  m    = mz[((size_t)b * CAPS + c) * 2 + 0];
    const float  invZ = 1.0f / mz[((size_t)b * CAPS + c) * 2 + 1];
    const float* xrow = x + (size_t)b * NODES * CIN;
    const float* lrow = logits + ((size_t)b * CAPS + c) * NODES;
    const float* Wc   = W + (size_t)c * NODES * CIN * COUT;

    v8f acc = {};                        // 16x16 f32 accumulator tile
    const int n0 = chunk * NPC;
#pragma unroll 2
    for (int n = n0; n < n0 + NPC; ++n) {
        const float wgt = __expf(lrow[n] - m) * invZ;   // c_ij on the fly
        // A 16x4 f32: lane holds (K = 2*half, 2*half+1) of row M=l16
        const float* xp = xrow + n * CIN + half * 2;
        v2f a_lo = { wgt * xp[0], wgt * xp[1] };        // K = 0..3
        v2f a_hi = { wgt * xp[4], wgt * xp[5] };        // K = 4..7
        // B 4x16 f32: lane holds rows (K = 2*half, 2*half+1), col N=l16
        const float* Wp = Wc + ((size_t)n * CIN + half * 2) * COUT + l16;
        v2f b_lo = { Wp[0],        Wp[COUT] };
        v2f b_hi = { Wp[4 * COUT], Wp[5 * COUT] };
        acc = __builtin_amdgcn_wmma_f32_16x16x4_f32(false, a_lo, false, b_lo,
                                                    (short)0, acc, false, false);
        acc = __builtin_amdgcn_wmma_f32_16x16x4_f32(false, a_hi, false, b_hi,
                                                    (short)0, acc, false, false);
    }
    // D layout: acc[r] = element (M = r + 8*half, N = l16)
#pragma unroll
    for (int r = 0; r < 8; ++r) {
        const int br = b0 + r + 8 * half;
        atomicAdd(&sbuf[((size_t)br * CAPS + c) * COUT + l16], acc[r]);
    }
}

// ---------------- squash: v = (|s|^2/(1+|s|^2)) * s/|s| ---------------------
__global__ void __launch_bounds__(256)
caps_squash_kernel(const float* __restrict__ sbuf, float* __restrict__ vbuf,
                   float* __restrict__ out, int write_out) {
    const int idx = blockIdx.x * 256 + threadIdx.x;   // over B*CAPS*COUT
    if (idx >= BATCH * CAPS * COUT) return;
    const int bc = idx / COUT;
    const float* sp = sbuf + (size_t)bc * COUT;
    float n2 = 0.0f;
#pragma unroll
    for (int o = 0; o < COUT; ++o) n2 += sp[o] * sp[o];
    const float scale = n2 / ((1.0f + n2) * sqrtf(n2));
    const float v = scale * sbuf[idx];
    vbuf[idx] = v;
    if (write_out) out[idx] = v;        // [B, caps, 1, out] contiguous
}

// ---------------- agreement: b_ij += <u_hat[b,c,n,:], v[b,c,:]> ------------
__global__ void __launch_bounds__(128)
caps_agree_kernel(const float* __restrict__ x, const float* __restrict__ W,
                  const float* __restrict__ vbuf, float* __restrict__ logits) {
    const int c     = blockIdx.x;
    const int chunk = blockIdx.y;
    const int wave  = threadIdx.x >> 5;
    const int lane  = threadIdx.x & 31;
    const int half  = lane >> 4;
    const int l16   = lane & 15;
    const int b0    = wave * 16;
    const int b     = b0 + l16;

    __shared__ float vsh[BATCH * COUT];              // v[:, c, :] (4 KB)
    for (int i = threadIdx.x; i < BATCH * COUT; i += 128) {
        const int bb = i / COUT, oo = i % COUT;
        vsh[i] = vbuf[((size_t)bb * CAPS + c) * COUT + oo];
    }
    __syncthreads();

    const float* xrow = x + (size_t)b * NODES * CIN;
    const float* Wc   = W + (size_t)c * NODES * CIN * COUT;
    const int n0 = chunk * NPC;
    for (int n = n0; n < n0 + NPC; ++n) {
        const float* xp = xrow + n * CIN + half * 2;
        v2f a_lo = { xp[0], xp[1] };
        v2f a_hi = { xp[4], xp[5] };
        const float* Wp = Wc + ((size_t)n * CIN + half * 2) * COUT + l16;
        v2f b_lo = { Wp[0],        Wp[COUT] };
        v2f b_hi = { Wp[4 * COUT], Wp[5 * COUT] };
        v8f acc = {};
        acc = __builtin_amdgcn_wmma_f32_16x16x4_f32(false, a_lo, false, b_lo,
                                                    (short)0, acc, false, false);
        acc = __builtin_amdgcn_wmma_f32_16x16x4_f32(false, a_hi, false, b_hi,
                                                    (short)0, acc, false, false);
        // acc[r] = u_hat[b = b0 + r + 8*half, n, o = l16]; reduce over o
#pragma unroll
        for (int r = 0; r < 8; ++r) {
            const int br = b0 + r + 8 * half;
            float t = acc[r] * vsh[br * COUT + l16];
            t += __shfl_xor(t, 1, 32);               // reductions stay within
            t += __shfl_xor(t, 2, 32);               // each 16-lane half
            t += __shfl_xor(t, 4, 32);
            t += __shfl_xor(t, 8, 32);
            if (l16 == 0) {                          // unique (b,c,n) writer
                const size_t idx = ((size_t)br * CAPS + c) * NODES + n;
                logits[idx] = logits[idx] + t;
            }
        }
    }
}

// ---------------- softmax stats over the 6272-node axis --------------------
__global__ void __launch_bounds__(256)
caps_softmax_kernel(const float* __restrict__ logits, float* __restrict__ mz) {
    const int bc = blockIdx.x;                       // b*CAPS + c
    const float* row = logits + (size_t)bc * NODES;
    __shared__ float red[256];
    float m = -1e30f;
    for (int n = threadIdx.x; n < NODES; n += 256) m = fmaxf(m, row[n]);
    red[threadIdx.x] = m;
    __syncthreads();
    for (int s = 128; s > 0; s >>= 1) {
        if (threadIdx.x < s)
            red[threadIdx.x] = fmaxf(red[threadIdx.x], red[threadIdx.x + s]);
        __syncthreads();
    }
    m = red[0];
    __syncthreads();
    float z = 0.0f;
    for (int n = threadIdx.x; n < NODES; n += 256) z += __expf(row[n] - m);
    red[threadIdx.x] = z;
    __syncthreads();
    for (int s = 128; s > 0; s >>= 1) {
        if (threadIdx.x < s) red[threadIdx.x] += red[threadIdx.x + s];
        __syncthreads();
    }
    if (threadIdx.x == 0) {
        mz[2 * bc + 0] = m;
        mz[2 * bc + 1] = red[0];
    }
}

// ---------------------------------------------------------------------------
extern "C" void kernel_launch(void* const* d_in, const int* in_sizes, int n_in,
                              void* d_out, int out_size, void* d_ws, size_t ws_size,
                              hipStream_t stream) {
    const float* x = (const float*)d_in[0];          // [64, 6272, 8]
    const float* W = (const float*)d_in[1];          // [10, 6272, 8, 16]
    float* out = (float*)d_out;                      // [64, 10, 1, 16]

    float* logits = (float*)d_ws;                                    // 4,014,080
    float* mz     = logits + (size_t)BATCH * CAPS * NODES;           // 1,280
    float* sbuf   = mz + (size_t)BATCH * CAPS * 2;                   // 10,240
    float* vbuf   = sbuf + (size_t)BATCH * CAPS * COUT;              // 10,240

    caps_init_kernel<<<2048, 256, 0, stream>>>(logits, mz, sbuf);

    const dim3 gemm_grid(CAPS, NCHUNK);              // 10 x 49 blocks, 128 thr
    for (int t = 0; t < 3; ++t) {
        caps_s_gemm_kernel<<<gemm_grid, 128, 0, stream>>>(x, W, logits, mz, sbuf);
        caps_squash_kernel<<<(BATCH * CAPS * COUT + 255) / 256, 256, 0, stream>>>(
            sbuf, vbuf, out, (t == 2) ? 1 : 0);
        if (t < 2) {
            caps_zero_s_kernel<<<(BATCH * CAPS * COUT + 255) / 256, 256, 0, stream>>>(sbuf);
            caps_agree_kernel<<<gemm_grid, 128, 0, stream>>>(x, W, vbuf, logits);
            caps_softmax_kernel<<<BATCH * CAPS, 256, 0, stream>>>(logits, mz);
        }
    }
}